// MonotonicRNNTLoss_70050916598250
// MI455X (gfx1250) — compile-verified
//
#include <hip/hip_runtime.h>
#include <hip/hip_bf16.h>

// CDNA5 / gfx1250 — wave32. WMMA f32 16x16x4 used as a 64-way f32 adder
// (B = all-ones) for the softmax denominator of the RNN-T log-softmax.
typedef __attribute__((ext_vector_type(2))) float v2f;
typedef __attribute__((ext_vector_type(8))) float v8f;

static constexpr int B_ = 8, T_ = 400, S_ = 40, V_ = 512;
static constexpr int U_ = S_ + 1;            // 41 states
static constexpr int NROWS = B_ * T_ * U_;   // 131200 softmax rows
static constexpr float NEGINF = -1e30f;

// ---------------------------------------------------------------------------
// Kernel 1: per-row log-sum-exp over V=512; emit lp_blank[row] and
// lp_label[b,t,u] (u<S). One wave32 per row; lane owns 16 contiguous floats
// loaded as 4x global_load_b128. Max via 5-step wave32 bpermute tree; exp-sum
// via 8 chained v_wmma_f32_16x16x4_f32 with an all-ones B operand.
//
// D/C layout (ISA 7.12.2): VGPR v, lanes 0-15 hold M=v (N=lane),
// lanes 16-31 hold M=v+8. With B = ones, every D row is its A-row group-sum
// replicated across N. Hence per-lane sum of the 8 D VGPRs is
//   lanes 0-15 : sum of rows 0-7,   lanes 16-31 : sum of rows 8-15,
// and the full exp-sum is one xor-16 shuffle combine — no /16 correction.
// ---------------------------------------------------------------------------
__global__ __launch_bounds__(256) void rnnt_lse_kernel(
    const float* __restrict__ acts,
    const int*   __restrict__ labels,
    float*       __restrict__ lp_blank,
    float*       __restrict__ lp_label) {
  const int lane = threadIdx.x & 31;
  const int wid  = threadIdx.x >> 5;
  const int row  = blockIdx.x * 8 + wid;
  if (row >= NROWS) return;   // uniform: NROWS == gridDim.x*8 exactly

  const float* base = acts + (size_t)row * V_;
  const float4* p = (const float4*)(base + lane * 16);
  float4 v0 = p[0];
  float4 v1 = p[1];
  float4 v2 = p[2];
  float4 v3 = p[3];

  // per-lane max of 16, then wave32 max reduction (5 xor-shuffles)
  float mx = fmaxf(fmaxf(fmaxf(v0.x, v0.y), fmaxf(v0.z, v0.w)),
             fmaxf(fmaxf(fmaxf(v1.x, v1.y), fmaxf(v1.z, v1.w)),
             fmaxf(fmaxf(fmaxf(v2.x, v2.y), fmaxf(v2.z, v2.w)),
                   fmaxf(fmaxf(v3.x, v3.y), fmaxf(v3.z, v3.w)))));
#pragma unroll
  for (int off = 16; off >= 1; off >>= 1)
    mx = fmaxf(mx, __shfl_xor(mx, off, 32));

  // exp(x - mx), packed into eight 16x4 f32 A fragments (2 VGPRs each).
  v2f a0 = { __expf(v0.x - mx), __expf(v0.y - mx) };
  v2f a1 = { __expf(v0.z - mx), __expf(v0.w - mx) };
  v2f a2 = { __expf(v1.x - mx), __expf(v1.y - mx) };
  v2f a3 = { __expf(v1.z - mx), __expf(v1.w - mx) };
  v2f a4 = { __expf(v2.x - mx), __expf(v2.y - mx) };
  v2f a5 = { __expf(v2.z - mx), __expf(v2.w - mx) };
  v2f a6 = { __expf(v3.x - mx), __expf(v3.y - mx) };
  v2f a7 = { __expf(v3.z - mx), __expf(v3.w - mx) };

  // D += A * ones. EXEC is all-ones here (no divergence before this point).
  const v2f ones = {1.0f, 1.0f};
  v8f c = {};
  c = __builtin_amdgcn_wmma_f32_16x16x4_f32(false, a0, false, ones, (short)0, c, false, false);
  c = __builtin_amdgcn_wmma_f32_16x16x4_f32(false, a1, false, ones, (short)0, c, false, false);
  c = __builtin_amdgcn_wmma_f32_16x16x4_f32(false, a2, false, ones, (short)0, c, false, false);
  c = __builtin_amdgcn_wmma_f32_16x16x4_f32(false, a3, false, ones, (short)0, c, false, false);
  c = __builtin_amdgcn_wmma_f32_16x16x4_f32(false, a4, false, ones, (short)0, c, false, false);
  c = __builtin_amdgcn_wmma_f32_16x16x4_f32(false, a5, false, ones, (short)0, c, false, false);
  c = __builtin_amdgcn_wmma_f32_16x16x4_f32(false, a6, false, ones, (short)0, c, false, false);
  c = __builtin_amdgcn_wmma_f32_16x16x4_f32(false, a7, false, ones, (short)0, c, false, false);

  // Half-wave row sums, then a single xor-16 combine for the full exp-sum.
  float s = (c[0] + c[1]) + (c[2] + c[3]) + ((c[4] + c[5]) + (c[6] + c[7]));
  s += __shfl_xor(s, 16, 32);

  const float logZ = mx + __logf(s);

  if (lane == 0) {
    // decode (b,t,u) from row = ((b*T + t)*(S+1) + u)
    const int b   = row / (T_ * U_);
    const int rem = row - b * (T_ * U_);
    const int t   = rem / U_;
    const int u   = rem - t * U_;
    lp_blank[row] = v0.x - logZ;                 // element 0 lives in lane 0
    if (u < S_) {
      const int lbl = labels[b * S_ + u];        // in [1, V)
      lp_label[(b * T_ + t) * S_ + u] = base[lbl] - logZ;  // L0/L2-hot reload
    }
  }
}

// ---------------------------------------------------------------------------
// Kernel 2: alpha recursion. One workgroup per batch element, 64 threads
// (2 waves) covering u = 0..40; LDS hands alpha[u-1] across the wave split.
// ---------------------------------------------------------------------------
__global__ __launch_bounds__(64) void rnnt_alpha_kernel(
    const float* __restrict__ lp_blank,
    const float* __restrict__ lp_label,
    const int*   __restrict__ input_lengths,
    const int*   __restrict__ label_lengths,
    float*       __restrict__ costs) {
  const int b = blockIdx.x;
  const int u = threadIdx.x;
  __shared__ float sh[U_ + 1];

  const int Tin  = input_lengths[b];
  const int Ulen = label_lengths[b];
  const float* lpb = lp_blank + (size_t)b * T_ * U_;
  const float* lpl = lp_label + (size_t)b * T_ * S_;

  float alpha = (u == 0) ? 0.0f : NEGINF;

  for (int t = 0; t < T_; ++t) {
    if (u <= S_) sh[u] = alpha;
    __syncthreads();
    if (u <= S_) {
      float nv = alpha + lpb[t * U_ + u];        // stay (emit blank)
      if (u >= 1) {
        const float move = sh[u - 1] + lpl[t * S_ + (u - 1)];
        const float hi = fmaxf(nv, move);
        const float lo = fminf(nv, move);
        nv = hi + log1pf(__expf(lo - hi));       // logaddexp
      }
      alpha = nv;
      if (t == Tin - 1 && u == Ulen) costs[b] = -nv;
    }
    __syncthreads();
  }
}

extern "C" void kernel_launch(void* const* d_in, const int* in_sizes, int n_in,
                              void* d_out, int out_size, void* d_ws, size_t ws_size,
                              hipStream_t stream) {
  const float* acts          = (const float*)d_in[0];
  const int*   labels        = (const int*)  d_in[1];
  const int*   input_lengths = (const int*)  d_in[2];
  const int*   label_lengths = (const int*)  d_in[3];
  float* out = (float*)d_out;

  float* lp_blank = (float*)d_ws;                    // B*T*(S+1) floats
  float* lp_label = lp_blank + (size_t)B_ * T_ * U_; // B*T*S floats

  rnnt_lse_kernel<<<NROWS / 8, 256, 0, stream>>>(acts, labels, lp_blank, lp_label);
  rnnt_alpha_kernel<<<B_, 64, 0, stream>>>(lp_blank, lp_label,
                                           input_lengths, label_lengths, out);
}